// SGConv_45518063403640
// MI455X (gfx1250) — compile-verified
//
#include <hip/hip_runtime.h>
#include <math.h>

// ---------------------------------------------------------------------------
// SGConv on MI455X (gfx1250, wave32):
//   deg/norm -> scatter hop1 (x norm[src]) -> scatter hop2 (x norm[src]^2)
//   -> WMMA f32 GEMM: out = (h * norm) @ W1^T + b1  using v_wmma_f32_16x16x4_f32
// ---------------------------------------------------------------------------

#define N_NODES 100000
#define N_EDGES 1600000
#define D_FEAT  64

typedef __attribute__((ext_vector_type(2))) float v2f;
typedef __attribute__((ext_vector_type(8))) float v8f;

// ---- degree / norm ---------------------------------------------------------

__global__ __launch_bounds__(256) void degree_kernel(const int* __restrict__ dst,
                                                     unsigned int* __restrict__ deg,
                                                     int nEdges) {
    int e = blockIdx.x * blockDim.x + threadIdx.x;
    if (e < nEdges) atomicAdd(&deg[dst[e]], 1u);
}

__global__ __launch_bounds__(256) void norm_kernel(const unsigned int* __restrict__ deg,
                                                   float* __restrict__ nrm, int n) {
    int i = blockIdx.x * blockDim.x + threadIdx.x;
    if (i < n) {
        float d = (float)deg[i];
        nrm[i] = rsqrtf(fmaxf(d, 1.0f));
    }
}

// ---- edge-parallel scatter: out[dst] += in[src] * norm[src]^(1 or 2) -------
// One wave (32 lanes) per edge; each lane moves 2 of the 64 features.

__global__ __launch_bounds__(256) void scatter_kernel(const float* __restrict__ in,
                                                      const float* __restrict__ nrm,
                                                      const int* __restrict__ src,
                                                      const int* __restrict__ dst,
                                                      float* __restrict__ out,
                                                      int nEdges, int squared) {
    const int lane = threadIdx.x & 31;
    const int e = blockIdx.x * 8 + (threadIdx.x >> 5);
    if (e >= nEdges) return;
    const int s = src[e];
    const int d = dst[e];
    float w = nrm[s];
    if (squared) w *= w;
    const float2 v = *reinterpret_cast<const float2*>(in + (size_t)s * D_FEAT + lane * 2);
    float* o = out + (size_t)d * D_FEAT + lane * 2;
    atomicAdd(o,     v.x * w);
    atomicAdd(o + 1, v.y * w);
}

// ---- WMMA GEMM: out[N,64] = (H * norm) @ W1^T + b1 -------------------------
// One wave computes one 16x16 output tile; K=64 consumed in 16 steps of K=4
// via v_wmma_f32_16x16x4_f32.
//
// A (16x4 f32) per ISA: lanes 0-15 hold M=0..15 with {VGPR0=K0, VGPR1=K1},
//                       lanes 16-31 hold M=0..15 with {VGPR0=K2, VGPR1=K3}.
// B (4x16 f32) mirrored over N. C/D: 8 VGPRs, lane%16 = N, M = vgpr + 8*(lane/16).

__global__ __launch_bounds__(256) void gemm_wmma_kernel(const float* __restrict__ H,
                                                        const float* __restrict__ nrm,
                                                        const float* __restrict__ W1,
                                                        const float* __restrict__ b1,
                                                        float* __restrict__ out,
                                                        int nNodes) {
    const int lane = threadIdx.x & 31;
    const int wave = threadIdx.x >> 5;
    const int tile = blockIdx.x * 8 + wave;       // 8 waves per block
    const int rowTile = tile >> 2;                // 4 col-tiles of 16 cover D_OUT=64
    const int colTile = tile & 3;
    const int m = rowTile * 16;
    const int n = colTile * 16;
    if (m >= nNodes) return;

    const int lo = lane & 15;                     // M (for A) / N (for B,C,D)
    const int hi = lane >> 4;                     // selects K pair {0,1} vs {2,3}

    const int arow = m + lo;
    const float nr = nrm[arow];                   // fold final *norm into A load
    const float bias = b1[n + lo];

    v8f acc;
#pragma unroll
    for (int r = 0; r < 8; ++r) acc[r] = bias;    // bias pre-loaded into C

    const float* __restrict__ hrow = H  + (size_t)arow * D_FEAT;
    const float* __restrict__ wrow = W1 + (size_t)(n + lo) * D_FEAT; // W1^T[k, n+lo]

#pragma unroll
    for (int k = 0; k < D_FEAT; k += 4) {
        const int c = k + 2 * hi;
        v2f a, b;
        a.x = hrow[c]     * nr;
        a.y = hrow[c + 1] * nr;
        b.x = wrow[c];
        b.y = wrow[c + 1];
        acc = __builtin_amdgcn_wmma_f32_16x16x4_f32(
            /*neg_a=*/false, a, /*neg_b=*/false, b,
            /*c_mod=*/(short)0, acc, /*reuse_a=*/false, /*reuse_b=*/false);
    }

#pragma unroll
    for (int r = 0; r < 8; ++r) {
        const int row = m + r + hi * 8;
        out[(size_t)row * D_FEAT + n + lo] = acc[r];
    }
}

// ---------------------------------------------------------------------------

extern "C" void kernel_launch(void* const* d_in, const int* in_sizes, int n_in,
                              void* d_out, int out_size, void* d_ws, size_t ws_size,
                              hipStream_t stream) {
    const float* feat = (const float*)d_in[0];
    const int*   src  = (const int*)d_in[1];
    const int*   dst  = (const int*)d_in[2];
    const float* W1   = (const float*)d_in[3];
    const float* b1   = (const float*)d_in[4];
    float*       out  = (float*)d_out;

    const int N = N_NODES;
    const int E = N_EDGES;

    // Workspace layout (bytes):
    //   [0,            512KB)  deg   (u32, 100000 used)
    //   [512KB,        1MB  )  norm  (f32, 100000 used)
    //   [1MB,          +25.6MB) tmpA (f32 N*64)
    //   [.. ,          +25.6MB) tmpB (f32 N*64)
    char* ws = (char*)d_ws;
    unsigned int* deg  = (unsigned int*)ws;
    float*        nrm  = (float*)(ws + (size_t)512 * 1024);
    float*        tmpA = (float*)(ws + (size_t)1024 * 1024);
    float*        tmpB = tmpA + (size_t)N * D_FEAT;
    const size_t  featBytes = (size_t)N * D_FEAT * sizeof(float);

    // 1) degrees + norm
    hipMemsetAsync(deg, 0, (size_t)N * sizeof(unsigned int), stream);
    degree_kernel<<<(E + 255) / 256, 256, 0, stream>>>(dst, deg, E);
    norm_kernel<<<(N + 255) / 256, 256, 0, stream>>>(deg, nrm, N);

    // 2) hop 1: tmpB[dst] += feat[src] * norm[src]
    hipMemsetAsync(tmpB, 0, featBytes, stream);
    scatter_kernel<<<(E + 7) / 8, 256, 0, stream>>>(feat, nrm, src, dst, tmpB, E, 0);

    // 3) hop 2: tmpA[dst] += tmpB[src] * norm[src]^2  (post-scale hop1 folded in)
    hipMemsetAsync(tmpA, 0, featBytes, stream);
    scatter_kernel<<<(E + 7) / 8, 256, 0, stream>>>(tmpB, nrm, src, dst, tmpA, E, 1);

    // 4) out = (tmpA * norm) @ W1^T + b1 via f32 WMMA
    const int rowTiles = (N + 15) / 16;           // 6250
    const int tiles = rowTiles * 4;               // 25000 wave-tiles
    gemm_wmma_kernel<<<(tiles + 7) / 8, 256, 0, stream>>>(tmpA, nrm, W1, b1, out, N);
}